// AttentiveMicroAggregator_54004918780323
// MI455X (gfx1250) — compile-verified
//
#include <hip/hip_runtime.h>
#include <math.h>

#define N_MICRO 500000
#define DIM     512
#define N_NODES 20000
#define N_EDGES 40000
#define N_SEG   60000
#define EPS_SM  1e-8f
#define EPS_LN  1e-5f
#define MAXE    2048
#define LDA     516   // padded LDS row stride (516 % 64 == 4 -> conflict-free A reads)
#define ROWS    32    // rows per k_proj block (two 16-row M-tiles)

typedef float v2f __attribute__((ext_vector_type(2)));
typedef float v8f __attribute__((ext_vector_type(8)));

// ---------------------------------------------------------------- K0: zero counts
__global__ void k_zero_counts(int* __restrict__ cnt) {
  int i = blockIdx.x * blockDim.x + threadIdx.x;
  if (i < N_SEG) cnt[i] = 0;
}

// ---------------------------------------------------------------- K1: logits + seg ids + histogram
// one wave32 per micro row; 8 waves per block
__global__ __launch_bounds__(256) void k_logits(
    const float* __restrict__ mf, const float* __restrict__ aw,
    const float* __restrict__ ab, const int* __restrict__ m2m,
    float* __restrict__ logits, int* __restrict__ seg, int* __restrict__ cnt)
{
  int wave = threadIdx.x >> 5;
  int lane = threadIdx.x & 31;
  int row  = blockIdx.x * 8 + wave;
  if (row >= N_MICRO) return;
  const float4* fr = (const float4*)(mf + (size_t)row * DIM);
  const float4* wr = (const float4*)aw;
  float acc = 0.f;
#pragma unroll
  for (int j = 0; j < 4; ++j) {
    float4 a = fr[lane * 4 + j];
    float4 b = wr[lane * 4 + j];
    acc += a.x * b.x + a.y * b.y + a.z * b.z + a.w * b.w;
  }
  for (int off = 16; off > 0; off >>= 1) acc += __shfl_down(acc, off, 32);
  if (lane == 0) {
    logits[row] = acc + ab[0];
    int m = m2m[row];
    int s = (m >= 0) ? m : ((m == -100) ? -1 : (N_NODES + (-m - 1)));
    seg[row] = s;
    if (s >= 0) atomicAdd(&cnt[s], 1);
  }
}

// ---------------------------------------------------------------- K2: exclusive prefix scan of counts
__global__ __launch_bounds__(1024) void k_scan(
    const int* __restrict__ cnt, int* __restrict__ off, int* __restrict__ cur)
{
  __shared__ int tmp[1024];
  __shared__ int carry_s;
  int t = threadIdx.x;
  if (t == 0) carry_s = 0;
  __syncthreads();
  for (int base = 0; base < N_SEG; base += 1024) {
    int v = (base + t < N_SEG) ? cnt[base + t] : 0;
    tmp[t] = v;
    __syncthreads();
    for (int d = 1; d < 1024; d <<= 1) {
      int add = (t >= d) ? tmp[t - d] : 0;
      __syncthreads();
      tmp[t] += add;
      __syncthreads();
    }
    int carry = carry_s;
    int excl  = carry + tmp[t] - v;
    if (base + t < N_SEG) { off[base + t] = excl; cur[base + t] = excl; }
    __syncthreads();
    if (t == 0) carry_s = carry + tmp[1023];
    __syncthreads();
  }
}

// ---------------------------------------------------------------- K3: scatter element ids per segment
__global__ void k_scatter(const int* __restrict__ seg, int* __restrict__ cur,
                          int* __restrict__ perm)
{
  int i = blockIdx.x * blockDim.x + threadIdx.x;
  if (i >= N_MICRO) return;
  int s = seg[i];
  if (s >= 0) { int p = atomicAdd(&cur[s], 1); perm[p] = i; }
}

// ---------------------------------------------------------------- K4: per-segment softmax + weighted feature sum
// one 256-thread block per segment; writes agg row directly into d_out
__global__ __launch_bounds__(256) void k_aggregate(
    const float* __restrict__ mf, const float* __restrict__ logits,
    const int* __restrict__ off, const int* __restrict__ cnt,
    const int* __restrict__ perm, float* __restrict__ agg)
{
  __shared__ int   sidx[MAXE];
  __shared__ float swt[MAXE];
  __shared__ float red[256];
  int s = blockIdx.x;
  int t = threadIdx.x;
  int o = off[s];
  int c = cnt[s];
  float2* orow = (float2*)(agg + (size_t)s * DIM);
  if (c == 0) { float2 z; z.x = 0.f; z.y = 0.f; orow[t] = z; return; }
  bool small = (c <= MAXE);
  if (small) {
    for (int j = t; j < c; j += 256) sidx[j] = perm[o + j];
    __syncthreads();
    if (t == 0) { // insertion sort -> deterministic accumulation order
      for (int a = 1; a < c; ++a) {
        int key = sidx[a]; int b = a - 1;
        while (b >= 0 && sidx[b] > key) { sidx[b + 1] = sidx[b]; --b; }
        sidx[b + 1] = key;
      }
    }
    __syncthreads();
  }
  // segment max
  float lm = -3.0e38f;
  for (int j = t; j < c; j += 256) {
    int id = small ? sidx[j] : perm[o + j];
    lm = fmaxf(lm, logits[id]);
  }
  red[t] = lm; __syncthreads();
  for (int d = 128; d > 0; d >>= 1) { if (t < d) red[t] = fmaxf(red[t], red[t + d]); __syncthreads(); }
  float mx = red[0];
  __syncthreads();
  // sum of exp
  float ls = 0.f;
  for (int j = t; j < c; j += 256) {
    int id = small ? sidx[j] : perm[o + j];
    float e = expf(logits[id] - mx);
    if (small) swt[j] = e;
    ls += e;
  }
  red[t] = ls; __syncthreads();
  for (int d = 128; d > 0; d >>= 1) { if (t < d) red[t] += red[t + d]; __syncthreads(); }
  float invden = 1.f / (red[0] + EPS_SM);
  __syncthreads();
  // weighted feature accumulation: thread t owns contiguous dims [2t, 2t+1]
  float ax = 0.f, ay = 0.f;
  for (int j = 0; j < c; ++j) {
    int id; float w;
    if (small) { id = sidx[j]; w = swt[j] * invden; }
    else       { id = perm[o + j]; w = expf(logits[id] - mx) * invden; }
    float2 v = ((const float2*)(mf + (size_t)id * DIM))[t];
    ax += v.x * w;
    ay += v.y * w;
  }
  float2 r; r.x = ax; r.y = ay;
  orow[t] = r;
}

// ---------------------------------------------------------------- K5: in-place [32,512]@[512,512] WMMA + LN + GELU
// 8 waves per block; wave w owns output columns [64w, 64w+64) for both 16-row M-tiles
__global__ __launch_bounds__(256) void k_proj(
    const float* __restrict__ W, const float* __restrict__ pb,
    const float* __restrict__ g, const float* __restrict__ lb,
    float* __restrict__ out)
{
  __shared__ float sA[ROWS * LDA];               // 66 KB
  __shared__ float psum[8][ROWS], psq[8][ROWS];
  __shared__ float smu[ROWS], srstd[ROWS];
  int t  = threadIdx.x;
  int r0 = blockIdx.x * ROWS;
  float* base = out + (size_t)r0 * DIM;

  // stage full 32x512 strip into padded LDS (read-before-overwrite makes in-place safe)
  for (int v = t; v < ROWS * DIM / 4; v += 256) {  // 4096 float4
    int row = v >> 7;
    int col = (v & 127) << 2;
    float4 x = ((const float4*)(base + row * DIM))[v & 127];
    sA[row * LDA + col + 0] = x.x;
    sA[row * LDA + col + 1] = x.y;
    sA[row * LDA + col + 2] = x.z;
    sA[row * LDA + col + 3] = x.w;
  }
  __syncthreads();

  int wave = t >> 5, lane = t & 31;
  int l16  = lane & 15;
  int koff = (lane >> 4) << 1;  // 0 for lanes 0-15, 2 for lanes 16-31
  int n0   = wave * 64;

  v8f acc[2][4] = {};
  const float* arow0 = &sA[l16 * LDA];
  const float* arow1 = &sA[(16 + l16) * LDA];
  for (int k0 = 0; k0 < DIM; k0 += 4) {
    v2f a0, a1;
    a0.x = arow0[k0 + koff];
    a0.y = arow0[k0 + koff + 1];
    a1.x = arow1[k0 + koff];
    a1.y = arow1[k0 + koff + 1];
    const float* w0 = W + (size_t)(k0 + koff) * DIM;
    const float* w1 = w0 + DIM;
#pragma unroll
    for (int tile = 0; tile < 4; ++tile) {
      int n = n0 + tile * 16 + l16;
      v2f b;
      b.x = w0[n];
      b.y = w1[n];
      acc[0][tile] = __builtin_amdgcn_wmma_f32_16x16x4_f32(
          false, a0, false, b, (short)0, acc[0][tile], false, false);
      acc[1][tile] = __builtin_amdgcn_wmma_f32_16x16x4_f32(
          false, a1, false, b, (short)0, acc[1][tile], false, false);
    }
  }

  // deterministic per-row LN stats: shfl within half-wave, fixed-order cross-wave sum
#pragma unroll
  for (int rt = 0; rt < 2; ++rt) {
#pragma unroll
    for (int j = 0; j < 8; ++j) {
      int R = rt * 16 + ((lane < 16) ? j : (j + 8));
      float s1 = 0.f, s2 = 0.f;
#pragma unroll
      for (int tile = 0; tile < 4; ++tile) {
        int n = n0 + tile * 16 + l16;
        float h = acc[rt][tile][j] + pb[n];
        s1 += h; s2 += h * h;
      }
      for (int m = 8; m > 0; m >>= 1) {
        s1 += __shfl_xor(s1, m, 16);
        s2 += __shfl_xor(s2, m, 16);
      }
      if (l16 == 0) { psum[wave][R] = s1; psq[wave][R] = s2; }
    }
  }
  __syncthreads();
  if (t < ROWS) {
    float s1 = 0.f, s2 = 0.f;
#pragma unroll
    for (int w = 0; w < 8; ++w) { s1 += psum[w][t]; s2 += psq[w][t]; }
    float mu  = s1 * (1.f / DIM);
    float var = s2 * (1.f / DIM) - mu * mu;
    smu[t]   = mu;
    srstd[t] = rsqrtf(var + EPS_LN);
  }
  __syncthreads();

  // LN + exact GELU + store (recompute h = acc + pb; acc stays live, no spill array)
#pragma unroll
  for (int rt = 0; rt < 2; ++rt) {
#pragma unroll
    for (int j = 0; j < 8; ++j) {
      int R = rt * 16 + ((lane < 16) ? j : (j + 8));
      float mu = smu[R], rs = srstd[R];
#pragma unroll
      for (int tile = 0; tile < 4; ++tile) {
        int n = n0 + tile * 16 + l16;
        float h = acc[rt][tile][j] + pb[n];
        float x = (h - mu) * rs * g[n] + lb[n];
        float y = 0.5f * x * (1.0f + erff(x * 0.70710678118654752f));
        base[R * DIM + n] = y;
      }
    }
  }
}

// ---------------------------------------------------------------- launch
extern "C" void kernel_launch(void* const* d_in, const int* in_sizes, int n_in,
                              void* d_out, int out_size, void* d_ws, size_t ws_size,
                              hipStream_t stream)
{
  const float* mf  = (const float*)d_in[0];  // micro_features [500000,512]
  const float* aw  = (const float*)d_in[1];  // attn_w [512]
  const float* ab  = (const float*)d_in[2];  // attn_b scalar
  const float* pw  = (const float*)d_in[3];  // proj_w [512,512]
  const float* pb  = (const float*)d_in[4];  // proj_b [512]
  const float* lg  = (const float*)d_in[5];  // ln_g [512]
  const float* lb  = (const float*)d_in[6];  // ln_b [512]
  const int*   m2m = (const int*)d_in[7];    // micro_to_macro [500000]
  float* out = (float*)d_out;                // [60000,512] node rows then edge rows

  char* ws = (char*)d_ws;
  float* logits = (float*)ws;  ws += sizeof(float) * N_MICRO;
  int*   seg    = (int*)ws;    ws += sizeof(int)   * N_MICRO;
  int*   perm   = (int*)ws;    ws += sizeof(int)   * N_MICRO;
  int*   cnt    = (int*)ws;    ws += sizeof(int)   * N_SEG;
  int*   off    = (int*)ws;    ws += sizeof(int)   * N_SEG;
  int*   cur    = (int*)ws;    ws += sizeof(int)   * N_SEG;

  k_zero_counts<<<(N_SEG + 255) / 256, 256, 0, stream>>>(cnt);
  k_logits<<<(N_MICRO + 7) / 8, 256, 0, stream>>>(mf, aw, ab, m2m, logits, seg, cnt);
  k_scan<<<1, 1024, 0, stream>>>(cnt, off, cur);
  k_scatter<<<(N_MICRO + 255) / 256, 256, 0, stream>>>(seg, cur, perm);
  k_aggregate<<<N_SEG, 256, 0, stream>>>(mf, logits, off, cnt, perm, out);
  k_proj<<<N_SEG / ROWS, 256, 0, stream>>>(pw, pb, lg, lb, out);
}